// GNNEncoder_73624329388567
// MI455X (gfx1250) — compile-verified
//
#include <hip/hip_runtime.h>

typedef float v2f __attribute__((ext_vector_type(2)));
typedef float v8f __attribute__((ext_vector_type(8)));

#define CDIM 128
#define HDIM 128

// ---------------- utility: grid-stride zero ----------------
__global__ void zero_f(float* __restrict__ p, int n) {
    int i = blockIdx.x * blockDim.x + threadIdx.x;
    int stride = gridDim.x * blockDim.x;
    for (; i < n; i += stride) p[i] = 0.0f;
}

// ---------------- degree accumulation ----------------
__global__ void degree_k(const long long* __restrict__ dst, float* __restrict__ deg, int E) {
    int e = blockIdx.x * blockDim.x + threadIdx.x;
    if (e < E) unsafeAtomicAdd(&deg[(int)dst[e]], 1.0f);
}

__global__ void invdeg_k(float* __restrict__ deg, int N) {
    int i = blockIdx.x * blockDim.x + threadIdx.x;
    if (i < N) deg[i] = 1.0f / fmaxf(deg[i], 1.0f);
}

// ---------------- edge scatter: one wave32 per edge ----------------
// lane loads float4 (32 lanes * 16B = full 128-float row), 4 float atomics/lane.
__global__ __launch_bounds__(256) void scatter_k(const float* __restrict__ x,
                                                 const long long* __restrict__ src,
                                                 const long long* __restrict__ dst,
                                                 float* __restrict__ agg, int E) {
    const int lane = threadIdx.x & 31;
    const int wave = threadIdx.x >> 5;
    const int e = blockIdx.x * 8 + wave;
    if (e >= E) return;
    // prefetch a future edge's source row (global_prefetch_b8 on gfx1250)
    if (e + 64 < E) {
        const float* pf = x + (size_t)((int)src[e + 64]) * CDIM + lane * 4;
        __builtin_prefetch(pf, 0, 1);
    }
    const int s = (int)src[e];
    const int d = (int)dst[e];
    const float4 v = ((const float4*)(x + (size_t)s * CDIM))[lane];
    float* ap = agg + (size_t)d * CDIM + lane * 4;
    unsafeAtomicAdd(ap + 0, v.x);
    unsafeAtomicAdd(ap + 1, v.y);
    unsafeAtomicAdd(ap + 2, v.z);
    unsafeAtomicAdd(ap + 3, v.w);
}

// ---------------- fused GEMM: pre = [agg*invdeg | x] @ [Wl;Wr] + b ----------------
// 256 threads = 8 waves; wave w handles rows [blk*128 + w*16, +16), all 128 output cols.
// K = 256 walked in 8 chunks of 32, each chunk staged in LDS (32x128 f32 = 16KB).
__global__ __launch_bounds__(256) void gemm_k(const float* __restrict__ agg,
                                              const float* __restrict__ invdeg,
                                              const float* __restrict__ xin,
                                              const float* __restrict__ Wl,
                                              const float* __restrict__ Wr,
                                              const float* __restrict__ bias,
                                              float* __restrict__ pre,
                                              float* __restrict__ gsum,
                                              float* __restrict__ gsum2, int N) {
    __shared__ float sW[32 * HDIM];  // 16 KB K-chunk of combined weights

    const int tid  = threadIdx.x;
    const int lane = tid & 31;
    const int wave = tid >> 5;
    const int half = lane >> 4;   // 0: K pair {0,1}; 1: K pair {2,3} (fp32 A layout)
    const int l16  = lane & 15;
    const int rbase = (blockIdx.x * 8 + wave) * 16;
    const int row  = rbase + l16;
    const int rowc = row < N ? row : N - 1;   // clamp loads; stores are guarded
    const float invd = invdeg[rowc];

    v8f acc[8];
#pragma unroll
    for (int j = 0; j < 8; ++j)
#pragma unroll
        for (int r = 0; r < 8; ++r) acc[j][r] = 0.0f;

    for (int kc = 0; kc < 8; ++kc) {
        const float* Wsrc = (kc < 4) ? (Wl + (size_t)kc * 32 * HDIM)
                                     : (Wr + (size_t)(kc - 4) * 32 * HDIM);
        __syncthreads();
        for (int i = tid; i < 32 * HDIM; i += 256) sW[i] = Wsrc[i];
        __syncthreads();

        // A source for this chunk (uniform branch across the wave)
        const float* abase;
        float ascale;
        if (kc < 4) { abase = agg + (size_t)rowc * CDIM + kc * 32;        ascale = invd; }
        else        { abase = xin + (size_t)rowc * CDIM + (kc - 4) * 32;  ascale = 1.0f; }

#pragma unroll
        for (int k = 0; k < 8; ++k) {
            const int kk = k * 4 + half * 2;
            v2f a;
            a.x = abase[kk + 0] * ascale;
            a.y = abase[kk + 1] * ascale;
#pragma unroll
            for (int j = 0; j < 8; ++j) {
                v2f bf;
                bf.x = sW[(kk + 0) * HDIM + j * 16 + l16];
                bf.y = sW[(kk + 1) * HDIM + j * 16 + l16];
                acc[j] = __builtin_amdgcn_wmma_f32_16x16x4_f32(
                    /*neg_a=*/false, a, /*neg_b=*/false, bf,
                    /*c_mod=*/(short)0, acc[j], /*reuse_a=*/false, /*reuse_b=*/false);
            }
        }
    }

    // epilogue: add bias, store pre-BN activations, accumulate BN statistics
#pragma unroll
    for (int j = 0; j < 8; ++j) {
        const int col = j * 16 + l16;
        const float bc = bias[col];
        float ls = 0.0f, ls2 = 0.0f;
#pragma unroll
        for (int r = 0; r < 8; ++r) {
            const int m = rbase + half * 8 + r;   // fp32 C/D layout
            const float val = acc[j][r] + bc;
            if (m < N) {
                pre[(size_t)m * HDIM + col] = val;
                ls  += val;
                ls2 += val * val;
            }
        }
        unsafeAtomicAdd(&gsum[col],  ls);
        unsafeAtomicAdd(&gsum2[col], ls2);
    }
}

// ---------------- BatchNorm finalize: scale/shift per channel ----------------
__global__ void bnfin_k(const float* __restrict__ gsum, const float* __restrict__ gsum2,
                        const float* __restrict__ gamma, const float* __restrict__ beta,
                        float* __restrict__ scale, float* __restrict__ shift, float invN) {
    const int c = threadIdx.x;  // 128 threads
    const float mu  = gsum[c] * invN;
    const float var = gsum2[c] * invN - mu * mu;
    const float s   = gamma[c] * rsqrtf(var + 1e-5f);
    scale[c] = s;
    shift[c] = beta[c] - mu * s;
}

// ---------------- apply BN + ReLU (in place over pre) ----------------
__global__ void apply_k(const float* __restrict__ pre, const float* __restrict__ scale,
                        const float* __restrict__ shift, float* __restrict__ out, int total) {
    int i = blockIdx.x * blockDim.x + threadIdx.x;
    int stride = gridDim.x * blockDim.x;
    for (; i < total; i += stride) {
        const int c = i & (HDIM - 1);
        const float v = pre[i] * scale[c] + shift[c];
        out[i] = fmaxf(v, 0.0f);
    }
}

extern "C" void kernel_launch(void* const* d_in, const int* in_sizes, int n_in,
                              void* d_out, int out_size, void* d_ws, size_t ws_size,
                              hipStream_t stream) {
    const float*     x     = (const float*)d_in[0];
    const long long* ei    = (const long long*)d_in[1];
    const float*     Wl    = (const float*)d_in[2];
    const float*     Wr    = (const float*)d_in[3];
    const float*     b     = (const float*)d_in[4];
    const float*     gamma = (const float*)d_in[5];
    const float*     beta  = (const float*)d_in[6];

    const int N = in_sizes[0] / CDIM;
    const int E = in_sizes[1] / 2;
    const long long* src = ei;
    const long long* dst = ei + E;

    float* ws = (float*)d_ws;
    size_t o = 0;
    float* deg   = ws + o; o += (size_t)N;
    float* agg   = ws + o; o += (size_t)N * CDIM;
    float* bufA  = ws + o; o += (size_t)N * CDIM;
    float* bufB  = ws + o; o += (size_t)N * CDIM;
    float* gsum  = ws + o; o += HDIM;
    float* gsum2 = ws + o; o += HDIM;
    float* scale = ws + o; o += HDIM;
    float* shift = ws + o; o += HDIM;

    const float* lin[3]  = { x, bufA, bufB };
    float*       lout[3] = { bufA, bufB, (float*)d_out };

    // degrees (shared by all layers)
    zero_f<<<(N + 255) / 256, 256, 0, stream>>>(deg, N);
    degree_k<<<(E + 255) / 256, 256, 0, stream>>>(dst, deg, E);
    invdeg_k<<<(N + 255) / 256, 256, 0, stream>>>(deg, N);

    const int aggN  = N * CDIM;
    const int gemmB = (N + 127) / 128;
    const int applyB = (aggN + 255) / 256;

    for (int l = 0; l < 3; ++l) {
        zero_f<<<(aggN + 255) / 256, 256, 0, stream>>>(agg, aggN);
        zero_f<<<1, 256, 0, stream>>>(gsum, 2 * HDIM);  // gsum + gsum2 contiguous
        scatter_k<<<(E + 7) / 8, 256, 0, stream>>>(lin[l], src, dst, agg, E);
        gemm_k<<<gemmB, 256, 0, stream>>>(agg, deg, lin[l],
                                          Wl + (size_t)l * CDIM * HDIM,
                                          Wr + (size_t)l * CDIM * HDIM,
                                          b + (size_t)l * HDIM,
                                          lout[l], gsum, gsum2, N);
        bnfin_k<<<1, HDIM, 0, stream>>>(gsum, gsum2,
                                        gamma + (size_t)l * HDIM,
                                        beta + (size_t)l * HDIM,
                                        scale, shift, 1.0f / (float)N);
        apply_k<<<applyB, 256, 0, stream>>>(lout[l], scale, shift, lout[l], aggN);
    }
}